// LQR_72378788872611
// MI455X (gfx1250) — compile-verified
//
#include <hip/hip_runtime.h>

// LQR batch solver for MI455X (gfx1250, wave32, WMMA).
//
// Factorization: Riccati matrices are batch-independent -> precompute per-t
//   G_t = E + K_t^T H        (9x9)   backward:  v' = G_t v + c_t,  k_t = L_t v
//   L_t = -Quu^-1 H          (3x9)
//   P_t = F_x + F_u K_t      (9x9)   forward:   x' = P_t x + F_u k_t
// stored directly as V_WMMA_F32_16X16X4_F32 A-fragments (full f32 precision).
// Main kernel: 1 wave = 16 batches kept as a 16x16 C-fragment (rows=components,
// cols=batches); 3 WMMAs per step; ds_swizzle SWAPX16 for C->B re-layout.

typedef float v2f __attribute__((ext_vector_type(2)));
typedef float v8f __attribute__((ext_vector_type(8)));

#define DT   0.1f
#define DT2  0.01f
#define DT3  0.001f
#define YAWW 10.0f
#define TF   80
#define TT   81

__device__ __forceinline__ float swapx16(float x) {
  // ds_swizzle group-of-32: and=0x1F, or=0, xor=0x10 -> swap lanes l <-> l^16
  int i = __builtin_bit_cast(int, x);
  i = __builtin_amdgcn_ds_swizzle(i, 0x401F);
  return __builtin_bit_cast(float, i);
}

// ---------------------------------------------------------------------------
// Kernel 1: serial Riccati sweep (batch-independent), single wave.
// Emits A-fragment tables into d_ws:
//   bwdTab[t in 0..80][lane 0..31][8]  : stacked [G_t ; L_t] (16x12 A-matrix)
//   fwdTab[t in 0..79][lane 0..31][8]  : P_t (16x12 A-matrix, rows 9..15 = 0)
// A-frag layout (f32 16x16x4): elem(M,K) -> lane = (M&15) + 16*(K/2), vgpr=K&1
// ---------------------------------------------------------------------------
__global__ __launch_bounds__(32) void lqr_riccati(float* __restrict__ ws) {
  const int tid = threadIdx.x;
  __shared__ float V[81], Vn[81], Q[144], W[9], K[27];

  const float dtc[4]    = {1.f, DT, DT2, DT3};
  const float wdiag[12] = {1.f, 1.f, YAWW, 0.f, 0.f, 1.f,
                           1.f, 1.f, 1.f,  0.1f, 0.1f, 0.1f};

  float* bwdTab = ws;
  float* fwdTab = ws + (size_t)TT * 256;

  for (int e = tid; e < 81; e += 32) V[e] = 0.f;
  __syncthreads();  // single-wave workgroup: barrier lowers to S_NOP

  for (int t = TF; t >= 0; --t) {
    // Q = C_t + F^T V F   (F banded: F[s, s+3d] = DT^d)
    for (int e = tid; e < 144; e += 32) {
      int i = e / 12, j = e % 12;
      float q = (t > 0 && i == j) ? wdiag[i] : 0.f;
      #pragma unroll
      for (int di = 0; di < 4; ++di) {
        int s = i - 3 * di;
        if (s < 0 || s > 8) continue;
        #pragma unroll
        for (int dj = 0; dj < 4; ++dj) {
          int u = j - 3 * dj;
          if (u < 0 || u > 8) continue;
          q += dtc[di] * dtc[dj] * V[s * 9 + u];
        }
      }
      Q[e] = q;
    }
    __syncthreads();

    if (tid == 0) {  // W = -Quu^{-1}  (3x3 adjugate)
      float a = Q[9*12+9],  b = Q[9*12+10],  c = Q[9*12+11];
      float d = Q[10*12+9], e0 = Q[10*12+10], f = Q[10*12+11];
      float g = Q[11*12+9], h = Q[11*12+10], i9 = Q[11*12+11];
      float A0 = e0*i9 - f*h, A1 = f*g - d*i9, A2 = d*h - e0*g;
      float s = -1.f / (a*A0 + b*A1 + c*A2);
      W[0] = s*A0; W[1] = s*(c*h - b*i9); W[2] = s*(b*f - c*e0);
      W[3] = s*A1; W[4] = s*(a*i9 - c*g); W[5] = s*(c*d - a*f);
      W[6] = s*A2; W[7] = s*(b*g - a*h);  W[8] = s*(a*e0 - b*d);
    }
    __syncthreads();

    // K = W * Qux  (3x9)   (K = -Quu^{-1} Qux)
    for (int e = tid; e < 27; e += 32) {
      int r = e / 9, cx = e % 9;
      K[e] = W[r*3+0]*Q[9*12+cx] + W[r*3+1]*Q[10*12+cx] + W[r*3+2]*Q[11*12+cx];
    }
    __syncthreads();

    // Vn = Qxx + Qxu K   (exact: Qux + Quu K == 0)
    for (int e = tid; e < 81; e += 32) {
      int i = e / 9, j = e % 9;
      Vn[e] = Q[i*12+j] + Q[i*12+9]*K[0*9+j] + Q[i*12+10]*K[1*9+j]
            + Q[i*12+11]*K[2*9+j];
    }

    // Emit A-fragments for this t.
    for (int e = tid; e < 256; e += 32) {
      int lp = e >> 3, idx = e & 7;
      float bval = 0.f, pval = 0.f;
      if (idx < 6) {
        int cH = idx >> 1, r = idx & 1;
        int M  = lp & 15;
        int cc = 4 * cH + 2 * (lp >> 4) + r;   // K-index of A element
        if (cc < 9) {
          float coefH = (cc < 3) ? DT3 : (cc < 6) ? DT2 : DT;  // H band coef
          int u = cc % 3;
          if (M < 9) {
            // G = E + K^T H ; E[i][s]: s==i ->1, s==i-3 ->DT, s==i-6 ->DT2
            float Ev = (cc == M) ? 1.f
                     : (M >= 3 && cc == M - 3) ? DT
                     : (M >= 6 && cc == M - 6) ? DT2 : 0.f;
            bval = Ev + coefH * K[u * 9 + M];
            // P = F_x + F_u K ; F_x[i][j]: j==i ->1, j==i+3 ->DT, j==i+6 ->DT2
            float Fx = (cc == M) ? 1.f
                     : (M < 6 && cc == M + 3) ? DT
                     : (M < 3 && cc == M + 6) ? DT2 : 0.f;
            float coefP = (M < 3) ? DT3 : (M < 6) ? DT2 : DT;  // F_u band coef
            pval = Fx + coefP * K[(M % 3) * 9 + cc];
          } else if (M < 12) {
            bval = coefH * W[(M - 9) * 3 + u];   // L = W H
          }
        }
      }
      bwdTab[((size_t)t * 32 + lp) * 8 + idx] = bval;
      if (t < TF) fwdTab[((size_t)t * 32 + lp) * 8 + idx] = pval;
    }
    __syncthreads();
    for (int e = tid; e < 81; e += 32) V[e] = Vn[e];
    __syncthreads();
  }
}

// ---------------------------------------------------------------------------
// Kernel 2: per-batch linear recursions via f32 WMMA.
// Block = 128 threads = 4 waves; wave handles 16 batches (one 16x16 tile).
// State C-fragment: row m (component) at vgpr m&7, lanes [16*(m>>3) .. +15],
// column n = batch within tile.
// ---------------------------------------------------------------------------
__global__ __launch_bounds__(128) void lqr_solve(
    const float* __restrict__ ego, const float* __restrict__ ap,
    const float* __restrict__ ws, float* __restrict__ out) {
  const int tid  = threadIdx.x;
  const int wave = tid >> 5, lane = tid & 31;
  const int col  = lane & 15, h = lane >> 4;
  const int b    = blockIdx.x * 64 + wave * 16 + col;

  const float* bwdTab = ws;
  const float* fwdTab = ws + (size_t)TT * 256;

  __shared__ float kst[4][TF * 16 * 3];   // 60 KB: per-wave k_t stash
  float* kmy = kst[wave];

  // ---------------- backward: v' = G v + c,  k = L v ----------------
  v8f acc = {};  // v_81 = 0
  for (int t = TF; t >= 0; --t) {
    const float* fp = bwdTab + ((size_t)t * 32 + lane) * 8;
    v2f A0 = { fp[0], fp[1] };
    v2f A1 = { fp[2], fp[3] };
    v2f A2 = { fp[4], fp[5] };

    v8f d = {};
    if (h == 0 && t > 0) {   // c_t rows 0..2 (targets, yaw scaled)
      const float* tp = ap + (((size_t)b * 6 + 5) * TF + (t - 1)) * 3;
      d[0] = -tp[0]; d[1] = -tp[1]; d[2] = -YAWW * tp[2];
    }

    // C-layout -> B-layout (4x16 per chunk): B[c][r] = v_row[4c + 2h + r]
    float sw0 = swapx16(acc[0]);
    float sw2 = swapx16(acc[2]);
    float sw3 = swapx16(acc[3]);
    float sw6 = swapx16(acc[6]);
    float sw7 = swapx16(acc[7]);
    v2f B0 = { h ? sw2 : acc[0], h ? sw3 : acc[1] };
    v2f B1 = { h ? sw6 : acc[4], h ? sw7 : acc[5] };
    v2f B2 = { h ? 0.f : sw0,    0.f };

    d = __builtin_amdgcn_wmma_f32_16x16x4_f32(false, A0, false, B0, (short)0, d, false, false);
    d = __builtin_amdgcn_wmma_f32_16x16x4_f32(false, A1, false, B1, (short)0, d, false, false);
    d = __builtin_amdgcn_wmma_f32_16x16x4_f32(false, A2, false, B2, (short)0, d, false, false);
    acc = d;   // rows 0..8 = v', rows 9..11 = k_t

    if (h == 1 && t < TF) {  // rows 9,10,11 live in vgprs 1,2,3 of half 1
      float* kp = kmy + ((size_t)t * 16 + col) * 3;
      kp[0] = acc[1]; kp[1] = acc[2]; kp[2] = acc[3];
    }
  }
  __syncthreads();

  // ---------------- forward: x' = P x + F_u k ----------------
  {
    const float* ep = ego + (size_t)b * 9;
    v8f x = {};
    if (h == 0) {
      x[0]=ep[0]; x[1]=ep[1]; x[2]=ep[2]; x[3]=ep[3];
      x[4]=ep[4]; x[5]=ep[5]; x[6]=ep[6]; x[7]=ep[7];
    } else {
      x[0]=ep[8];
    }
    acc = x;
  }
  for (int t = 0; t < TF; ++t) {
    const float* fp = fwdTab + ((size_t)t * 32 + lane) * 8;
    v2f A0 = { fp[0], fp[1] };
    v2f A1 = { fp[2], fp[3] };
    v2f A2 = { fp[4], fp[5] };

    const float* kp = kmy + ((size_t)t * 16 + col) * 3;
    float k0 = kp[0], k1 = kp[1], k2 = kp[2];
    v8f d = {};
    if (h == 0) {            // F_u k: rows 0..2 DT^3, 3..5 DT^2, 6..8 DT
      d[0] = DT3*k0; d[1] = DT3*k1; d[2] = DT3*k2;
      d[3] = DT2*k0; d[4] = DT2*k1; d[5] = DT2*k2;
      d[6] = DT *k0; d[7] = DT *k1;
    } else {
      d[0] = DT * k2;        // row 8
    }

    float sw0 = swapx16(acc[0]);
    float sw2 = swapx16(acc[2]);
    float sw3 = swapx16(acc[3]);
    float sw6 = swapx16(acc[6]);
    float sw7 = swapx16(acc[7]);
    v2f B0 = { h ? sw2 : acc[0], h ? sw3 : acc[1] };
    v2f B1 = { h ? sw6 : acc[4], h ? sw7 : acc[5] };
    v2f B2 = { h ? 0.f : sw0,    0.f };

    d = __builtin_amdgcn_wmma_f32_16x16x4_f32(false, A0, false, B0, (short)0, d, false, false);
    d = __builtin_amdgcn_wmma_f32_16x16x4_f32(false, A1, false, B1, (short)0, d, false, false);
    d = __builtin_amdgcn_wmma_f32_16x16x4_f32(false, A2, false, B2, (short)0, d, false, false);
    acc = d;   // rows 0..8 = x_{t+1}

    if (h == 0) {            // plan[b][t][0..2] = x_{t+1}[0..2]
      float* op = out + ((size_t)b * TF + t) * 3;
      op[0] = acc[0]; op[1] = acc[1]; op[2] = acc[2];
    }
  }
}

extern "C" void kernel_launch(void* const* d_in, const int* in_sizes, int n_in,
                              void* d_out, int out_size, void* d_ws, size_t ws_size,
                              hipStream_t stream) {
  const float* ego = (const float*)d_in[0];   // (B, 9) f32
  const float* ap  = (const float*)d_in[1];   // (B, 6, 80, 3) f32
  float* out = (float*)d_out;                 // (B, 80, 3) f32
  float* ws  = (float*)d_ws;                  // needs (81+80)*256*4 = ~161 KB

  const int B = in_sizes[0] / 9;              // 8192
  lqr_riccati<<<1, 32, 0, stream>>>(ws);
  lqr_solve<<<B / 64, 128, 0, stream>>>(ego, ap, ws, out);
}